// LSHGaussian_62723702391547
// MI455X (gfx1250) — compile-verified
//
#include <hip/hip_runtime.h>
#include <hip/hip_bf16.h>

// ---------------------------------------------------------------------------
// LSH Gaussian filter (E2LSH, L=5 tables, K=5 projections, W=30, 2^18 buckets)
//
// Pipeline:
//   1) lsh_ids_wmma : codes = floor(ref x (A/W)^T + b) via
//      V_WMMA_F32_16X16X4_F32 (1/W folded into the B operand: algebraically
//      identical to floor((ref.A + b*W)/W), no per-element divide needed),
//      ids = (sum codes*primes) & (2^18-1)  -> ws
//   2) per table l: memset bucket sums/counts, scatter U with f32 atomics
//      (table is L2-resident: 33 MB << 192 MB L2), gather-accumulate into out
//   3) finalize: out = out/5 - U
// ---------------------------------------------------------------------------

#define LSH_C 32
#define LSH_D 16
#define LSH_L 5
#define LSH_K 5
#define LSH_J 25            // L*K columns of the projection GEMM
#define LSH_W 30.0f
#define LSH_NB (1 << 18)
#define LSH_MASK (LSH_NB - 1)

typedef __attribute__((ext_vector_type(2))) float v2f;
typedef __attribute__((ext_vector_type(8))) float v8f;

// ---------------------------------------------------------------------------
// Kernel 1: per-wave 16-point tile, 16x16x4 f32 WMMA, 4 K-steps x 2 N-tiles.
//
// A-operand (16x4 f32): lane<16 -> M=lane, K={0,1}; lane>=16 -> M=lane-16,
// K={2,3}.  B-operand (4x16 f32): same pattern with N striped across lanes.
// C/D (16x16 f32, 8 VGPRs): VGPR r holds M=r (lanes 0-15) / M=r+8 (16-31).
// ---------------------------------------------------------------------------
__global__ __launch_bounds__(256)
void lsh_ids_wmma(const float* __restrict__ ref,   // [N][16]
                  const float* __restrict__ Amat,  // [25][16] (L,K,D flattened)
                  const float* __restrict__ bvec,  // [25]
                  int* __restrict__ ids,           // [5][N]
                  int N)
{
    __shared__ int codes_lds[8][16][32];   // [wave][point-in-tile][column]

    const int lane  = threadIdx.x & 31;
    const int wave  = threadIdx.x >> 5;
    const int nIdx  = lane & 15;           // column / row-in-tile sub-index
    const int khalf = lane >> 4;           // 0 -> K pair {0,1}, 1 -> {2,3}
    const int tileBase = (blockIdx.x * 8 + wave) * 16;
    const bool active = tileBase < N;      // uniform per wave -> EXEC all-1s

    // B operand: rows of the projection matrix, pre-scaled by 1/W, for two
    // 16-column tiles (constant for the whole kernel; cols >= 25 are zero).
    const float invW = 1.0f / LSH_W;
    v2f bop[2][4];
#pragma unroll
    for (int t = 0; t < 2; ++t) {
        const int col = t * 16 + nIdx;
        const bool v  = col < LSH_J;
#pragma unroll
        for (int s = 0; s < 4; ++s) {
            const int d0 = 4 * s + 2 * khalf;
            bop[t][s].x = v ? Amat[col * LSH_D + d0]     * invW : 0.0f;
            bop[t][s].y = v ? Amat[col * LSH_D + d0 + 1] * invW : 0.0f;
        }
    }
    // floor((ref.A + b*W)/W) == floor(ref.(A/W) + b): bias is just b.
    const float bias0 = (nIdx      < LSH_J) ? bvec[nIdx]      : 0.0f;
    const float bias1 = (16 + nIdx < LSH_J) ? bvec[16 + nIdx] : 0.0f;

    if (active) {
        int point = tileBase + nIdx;       // N % 16 == 0 -> always full tile
        if (point >= N) point = N - 1;     // clamp (safety only)

        // A operand: this lane's two ref features per K-step.
        const float4* r4 = reinterpret_cast<const float4*>(ref + (size_t)point * LSH_D);
        v2f aop[4];
#pragma unroll
        for (int s = 0; s < 4; ++s) {
            const float4 q = r4[s];
            aop[s].x = khalf ? q.z : q.x;
            aop[s].y = khalf ? q.w : q.y;
        }

        v8f acc0 = {};   // columns 0..15
        v8f acc1 = {};   // columns 16..31 (16..24 used)
#pragma unroll
        for (int s = 0; s < 4; ++s) {
            acc0 = __builtin_amdgcn_wmma_f32_16x16x4_f32(
                false, aop[s], false, bop[0][s], (short)0, acc0, false, false);
            acc1 = __builtin_amdgcn_wmma_f32_16x16x4_f32(
                false, aop[s], false, bop[1][s], (short)0, acc1, false, false);
        }

        // Quantize to E2LSH codes and park in LDS for the per-point hash.
#pragma unroll
        for (int r = 0; r < 8; ++r) {
            const int m = r + 8 * khalf;   // point-in-tile this element maps to
            codes_lds[wave][m][nIdx]      = (int)floorf(acc0[r] + bias0);
            codes_lds[wave][m][16 + nIdx] = (int)floorf(acc1[r] + bias1);
        }
    }

    __syncthreads();

    // Lanes 0..15: one point each -> 5 bucket ids.  unsigned arithmetic +
    // mask reproduces JAX's wrapping int32 mul-sum + floor-mod(2^18) exactly.
    if (active && lane < 16) {
        const int point = tileBase + lane;
        if (point < N) {
            const unsigned P[LSH_K] = {73856093u, 19349663u, 83492791u,
                                       49979687u, 86028157u};
#pragma unroll
            for (int l = 0; l < LSH_L; ++l) {
                unsigned h = 0;
#pragma unroll
                for (int k = 0; k < LSH_K; ++k)
                    h += (unsigned)codes_lds[wave][lane][l * LSH_K + k] * P[k];
                ids[(size_t)l * N + point] = (int)(h & LSH_MASK);
            }
        }
    }
}

// ---------------------------------------------------------------------------
// Kernel 2: scatter U into per-bucket sums + counts (f32 atomics, L2-resident)
// One thread per (point, channel) pair -> coalesced U reads.
// ---------------------------------------------------------------------------
__global__ __launch_bounds__(256)
void lsh_scatter(const float* __restrict__ U,      // [N][32]
                 const int* __restrict__ idsl,     // [N] (one table)
                 float* __restrict__ bsum,         // [NB][32]
                 float* __restrict__ bcnt,         // [NB]
                 long long total)                  // N*32
{
    const long long i = (long long)blockIdx.x * blockDim.x + threadIdx.x;
    if (i >= total) return;
    const int n  = (int)(i >> 5);
    const int c  = (int)(i & 31);
    const int id = idsl[n];
    atomicAdd(&bsum[(size_t)id * LSH_C + c], U[i]);
    if (c == 0) atomicAdd(&bcnt[id], 1.0f);
}

// ---------------------------------------------------------------------------
// Kernel 3: gather bucket mean back to each point, accumulate into out.
// Precise f32 divide kept here to match the reference's s[id]/c exactly.
// ---------------------------------------------------------------------------
__global__ __launch_bounds__(256)
void lsh_gather(const int* __restrict__ idsl,
                const float* __restrict__ bsum,
                const float* __restrict__ bcnt,
                float* __restrict__ out,           // [N][32] accumulator
                long long total)
{
    const long long i = (long long)blockIdx.x * blockDim.x + threadIdx.x;
    if (i >= total) return;
    const int n  = (int)(i >> 5);
    const int c  = (int)(i & 31);
    const int id = idsl[n];
    const float cnt = bcnt[id];
    out[i] += bsum[(size_t)id * LSH_C + c] / fmaxf(cnt, 1.0f);
}

// ---------------------------------------------------------------------------
// Kernel 4: out = acc/L - U
// ---------------------------------------------------------------------------
__global__ __launch_bounds__(256)
void lsh_finalize(const float* __restrict__ U,
                  float* __restrict__ out,
                  long long total)
{
    const long long i = (long long)blockIdx.x * blockDim.x + threadIdx.x;
    if (i >= total) return;
    out[i] = out[i] * (1.0f / (float)LSH_L) - U[i];
}

// ---------------------------------------------------------------------------
extern "C" void kernel_launch(void* const* d_in, const int* in_sizes, int n_in,
                              void* d_out, int out_size, void* d_ws, size_t ws_size,
                              hipStream_t stream)
{
    (void)n_in; (void)out_size; (void)ws_size;

    const float* U    = (const float*)d_in[0];   // [N][32]
    const float* ref  = (const float*)d_in[1];   // [N][16]
    const float* Amat = (const float*)d_in[2];   // [5][5][16]
    const float* bvec = (const float*)d_in[3];   // [5][5]
    float*       out  = (float*)d_out;

    const int N = in_sizes[0] / LSH_C;
    const long long total = (long long)N * LSH_C;

    // Workspace layout: ids[5][N] | bsum[NB][32] | bcnt[NB]   (~43 MB)
    char*  ws   = (char*)d_ws;
    int*   ids  = (int*)ws;
    size_t off  = (((size_t)LSH_L * N * sizeof(int)) + 255) & ~(size_t)255;
    float* bsum = (float*)(ws + off);
    off += (size_t)LSH_NB * LSH_C * sizeof(float);
    float* bcnt = (float*)(ws + off);

    hipMemsetAsync(out, 0, (size_t)total * sizeof(float), stream);

    const dim3 blk(256);
    const int gridIds = (N + 127) / 128;                 // 8 waves x 16 pts
    lsh_ids_wmma<<<gridIds, blk, 0, stream>>>(ref, Amat, bvec, ids, N);

    const int gridNC = (int)((total + 255) / 256);
    for (int l = 0; l < LSH_L; ++l) {
        hipMemsetAsync(bsum, 0, (size_t)LSH_NB * LSH_C * sizeof(float), stream);
        hipMemsetAsync(bcnt, 0, (size_t)LSH_NB * sizeof(float), stream);
        lsh_scatter<<<gridNC, blk, 0, stream>>>(U, ids + (size_t)l * N,
                                                bsum, bcnt, total);
        lsh_gather<<<gridNC, blk, 0, stream>>>(ids + (size_t)l * N,
                                               bsum, bcnt, out, total);
    }
    lsh_finalize<<<gridNC, blk, 0, stream>>>(U, out, total);
}